// TripletCenterLoss_1769526526580
// MI455X (gfx1250) — compile-verified
//
#include <hip/hip_runtime.h>
#include <hip/hip_bf16.h>
#include <math.h>

typedef __attribute__((ext_vector_type(2))) float v2f;
typedef __attribute__((ext_vector_type(8))) float v8f;

#define BN   8192          // batch
#define DK   512           // feature dim
#define NC   55            // classes
#define NCP  64            // padded classes (4 N-tiles of 16)
#define KC   128           // K chunk staged in LDS
#define LSTR (KC + 4)      // LDS row stride (floats): 132 -> bank = 4n mod 64, conflict-free
#define MARGIN_F 0.2f

// ---------------- prep: class presence + ||center_c||^2 (deterministic) ----------------
__global__ __launch_bounds__(256) void tcl_prep(const float* __restrict__ centers,
                                                const int* __restrict__ targets,
                                                float* __restrict__ c2,
                                                float* __restrict__ present) {
    __shared__ int pl[NCP];
    int tid = threadIdx.x;
    if (tid < NCP) pl[tid] = 0;
    __syncthreads();
    for (int j = tid; j < BN; j += 256) pl[targets[j]] = 1;   // all writers store 1: deterministic
    __syncthreads();
    if (tid < NCP) {
        float s = 0.0f;
        if (tid < NC) {
            const float* row = centers + tid * DK;
            for (int k = 0; k < DK; ++k) s += row[k] * row[k]; // fixed order: deterministic
        }
        c2[tid]      = s;
        present[tid] = (float)pl[tid];
    }
}

// ---------------- main: WMMA GEMM (inputs @ centers^T) fused with loss epilogue --------
__global__ __launch_bounds__(256) void tcl_main(const float* __restrict__ inputs,
                                                const int* __restrict__ targets,
                                                const float* __restrict__ centers,
                                                const float* __restrict__ c2g,
                                                const float* __restrict__ presg,
                                                float* __restrict__ lossPart,
                                                float* __restrict__ precPart) {
    __shared__ float sB[NCP * LSTR];   // 64 x 132 floats = 33 KB
    __shared__ float wl[8], wp[8];

    const int tid  = threadIdx.x;
    const int lane = tid & 31;
    const int wave = tid >> 5;
    const int half = lane >> 4;        // 0: K{0,1} / rows 0..7 of C ; 1: K{2,3} / rows 8..15
    const int l16  = lane & 15;
    const int rowBase = blockIdx.x * 128 + wave * 16;

    v8f acc[4];
    acc[0] = v8f{}; acc[1] = v8f{}; acc[2] = v8f{}; acc[3] = v8f{};
    float x2p = 0.0f;                  // partial ||x_row||^2 (this half's K positions)

    for (int kc = 0; kc < DK; kc += KC) {
        __syncthreads();               // previous chunk fully consumed
        // stage centers[*, kc:kc+KC] into LDS (rows >= NC zero-padded)
        for (int e = tid; e < NCP * KC; e += 256) {
            int r = e >> 7, c = e & (KC - 1);
            sB[r * LSTR + c] = (r < NC) ? centers[r * DK + kc + c] : 0.0f;
        }
        __syncthreads();

        const float* aRow  = inputs + (size_t)(rowBase + l16) * DK + kc + half * 2;
        const float* bBase = &sB[l16 * LSTR + half * 2];
#pragma unroll 4
        for (int k = 0; k < KC; k += 4) {
            v2f a  = *(const v2f*)(aRow + k);
            v2f b0 = *(const v2f*)(bBase + k);
            v2f b1 = *(const v2f*)(bBase + k + 16 * LSTR);
            v2f b2 = *(const v2f*)(bBase + k + 32 * LSTR);
            v2f b3 = *(const v2f*)(bBase + k + 48 * LSTR);
            x2p += a[0] * a[0] + a[1] * a[1];
            acc[0] = __builtin_amdgcn_wmma_f32_16x16x4_f32(false, a, false, b0, (short)0, acc[0], false, false);
            acc[1] = __builtin_amdgcn_wmma_f32_16x16x4_f32(false, a, false, b1, (short)0, acc[1], false, false);
            acc[2] = __builtin_amdgcn_wmma_f32_16x16x4_f32(false, a, false, b2, (short)0, acc[2], false, false);
            acc[3] = __builtin_amdgcn_wmma_f32_16x16x4_f32(false, a, false, b3, (short)0, acc[3], false, false);
        }
    }

    // complete ||x||^2 per row: lane l16 (both halves) now holds row rowBase+l16
    float x2full = x2p + __shfl_xor(x2p, 16, 32);

    float c2v[4], prs[4];
#pragma unroll
    for (int t = 0; t < 4; ++t) {
        int c = l16 + 16 * t;
        c2v[t] = c2g[c];
        prs[t] = presg[c];
    }

    const float NINF = -__builtin_huge_valf();
    const float PINF =  __builtin_huge_valf();
    float lsum = 0.0f, psum = 0.0f;

#pragma unroll
    for (int v = 0; v < 8; ++v) {
        int   rloc = v + 8 * half;                    // local row this half owns in VGPR v
        int   trow = targets[rowBase + rloc];
        float x2r  = __shfl(x2full, rloc, 32);        // broadcast row norm
        float qa = NINF, qn = PINF;                   // q = c^2 - 2*G ; d = sqrt(x^2 + q)
#pragma unroll
        for (int t = 0; t < 4; ++t) {
            int   c = l16 + 16 * t;
            float q = c2v[t] - 2.0f * acc[t][v];
            if (c == trow) qa = q;                                    // hardest positive (unique)
            qn = (prs[t] != 0.0f && c != trow) ? fminf(qn, q) : qn;   // hardest negative
        }
#pragma unroll
        for (int ofs = 1; ofs < 16; ofs <<= 1) {      // reduce across the 16 column-lanes
            qa = fmaxf(qa, __shfl_xor(qa, ofs, 32));
            qn = fminf(qn, __shfl_xor(qn, ofs, 32));
        }
        float ap = sqrtf(fmaxf(x2r + qa, 1e-12f));
        float an = sqrtf(fmaxf(x2r + qn, 1e-12f));
        if (l16 == 0) {                               // one lane per half contributes per row
            lsum += fmaxf(ap - an + MARGIN_F, 0.0f);
            psum += (an > ap) ? 1.0f : 0.0f;
        }
    }
    lsum += __shfl_xor(lsum, 16, 32);
    psum += __shfl_xor(psum, 16, 32);
    if (lane == 0) { wl[wave] = lsum; wp[wave] = psum; }
    __syncthreads();
    if (tid == 0) {
        float L = 0.0f, P = 0.0f;
#pragma unroll
        for (int i = 0; i < 8; ++i) { L += wl[i]; P += wp[i]; }
        lossPart[blockIdx.x] = L;
        precPart[blockIdx.x] = P;
    }
}

// ---------------- finish: fixed-order final reduction (deterministic) ------------------
__global__ void tcl_finish(const float* __restrict__ lossPart,
                           const float* __restrict__ precPart,
                           float* __restrict__ out) {
    if (threadIdx.x == 0 && blockIdx.x == 0) {
        float l = 0.0f, p = 0.0f;
        for (int i = 0; i < 64; ++i) { l += lossPart[i]; p += precPart[i]; }
        out[0] = l * (1.0f / (float)BN);
        out[1] = p * (1.0f / (float)BN);
    }
}

extern "C" void kernel_launch(void* const* d_in, const int* in_sizes, int n_in,
                              void* d_out, int out_size, void* d_ws, size_t ws_size,
                              hipStream_t stream) {
    const float* inputs  = (const float*)d_in[0];
    const int*   targets = (const int*)d_in[1];
    const float* centers = (const float*)d_in[2];
    float* wsf      = (float*)d_ws;
    float* c2       = wsf;          // [64]
    float* present  = wsf + 64;     // [64]
    float* lossPart = wsf + 128;    // [64]
    float* precPart = wsf + 192;    // [64]

    tcl_prep  <<<1, 256, 0, stream>>>(centers, targets, c2, present);
    tcl_main  <<<64, 256, 0, stream>>>(inputs, targets, centers, c2, present, lossPart, precPart);
    tcl_finish<<<1, 32, 0, stream>>>(lossPart, precPart, (float*)d_out);
}